// TripletHardLoss_81767587381277
// MI455X (gfx1250) — compile-verified
//
#include <hip/hip_runtime.h>
#include <math.h>

typedef __attribute__((ext_vector_type(2))) float v2f;
typedef __attribute__((ext_vector_type(8))) float v8f;

#define N_EMB 8192
#define DIM 128
#define KSTEPS (DIM / 4)               // 32 wmma f32 16x16x4 steps per tile
#define CHUNKS 4
#define ROWBLKS (N_EMB / 128)          // 64 row-blocks of 128 rows
#define COLS_PER_CHUNK (N_EMB / CHUNKS)          // 2048
#define TILES_PER_CHUNK (COLS_PER_CHUNK / 16)    // 128
#define LDS_STRIDE 132                 // 128 + 4 pad -> conflict-free b64 reads
#define MARGIN_F 1.0f
#define EPS_F 1e-6f

// ---------------- Kernel 1: row squared norms ----------------
__global__ __launch_bounds__(256) void tl_rownorm(const float* __restrict__ E,
                                                  float* __restrict__ sq) {
  const int wave = threadIdx.x >> 5;
  const int lane = threadIdx.x & 31;
  const int row = blockIdx.x * 8 + wave;
  const float4* p = (const float4*)(E + (size_t)row * DIM);
  float4 v = p[lane];
  float s = v.x * v.x + v.y * v.y + v.z * v.z + v.w * v.w;
#pragma unroll
  for (int m = 1; m < 32; m <<= 1) s += __shfl_xor(s, m, 32);
  if (lane == 0) sq[row] = s;
}

// ---------------- Kernel 2: blocked fused Gram GEMM + hardest pos/neg ------
// Block: 256 threads = 8 waves. Block owns 128 anchor rows (wave w: 16-row
// strip). All waves share each 16-column B tile staged through LDS (8x reuse).
// Grid: 64 row-blocks x 4 column chunks.
__global__ __launch_bounds__(256) void tl_gram_hard(
    const float* __restrict__ E, const int* __restrict__ labels,
    const float* __restrict__ sq, float* __restrict__ cpv,
    int* __restrict__ cpi, float* __restrict__ cnv, int* __restrict__ cni) {
  __shared__ float Bt[2][16][LDS_STRIDE];

  const int tid = threadIdx.x;
  const int wave = tid >> 5;
  const int lane = tid & 31;
  const int l16 = lane & 15;
  const int half = lane >> 4;  // 0: acc rows i, 1: acc rows i+8
  const int koff = half * 2;   // A/B fragment k sub-offset
  const int rowBlk = blockIdx.x & (ROWBLKS - 1);
  const int chunk = blockIdx.x / ROWBLKS;
  const int rowBase = rowBlk * 128 + wave * 16;
  const int colChunk0 = chunk * COLS_PER_CHUNK;

  // Preload A fragments for this wave's 16 anchor rows.
  const float* arow = E + (size_t)(rowBase + l16) * DIM + koff;
  v2f aF[KSTEPS];
#pragma unroll
  for (int t = 0; t < KSTEPS; ++t) {
    aF[t].x = arow[4 * t];
    aF[t].y = arow[4 * t + 1];
  }

  // Per accumulator-slot metadata: slot i -> row rowBase + i + 8*half.
  float sqA[8];
  int labA[8];
#pragma unroll
  for (int i = 0; i < 8; ++i) {
    const int r = rowBase + i + 8 * half;
    sqA[i] = sq[r];
    labA[i] = labels[r];
  }

  float posv[8], negv[8];
  int posi[8], negi[8];
#pragma unroll
  for (int i = 0; i < 8; ++i) {
    posv[i] = -INFINITY;
    negv[i] = INFINITY;
    posi[i] = 0;
    negi[i] = 0;
  }

  // Cooperative staging: thread -> (tile row, 8-float segment).
  const int stRow = tid >> 4;  // 0..15 (column within tile)
  const int stSeg = tid & 15;  // 0..15 (two float4 = 32B)

  // Stage tile 0.
  {
    const float4* g = (const float4*)(E +
        (size_t)(colChunk0 + stRow) * DIM + stSeg * 8);
    float4 r0 = g[0], r1 = g[1];
    float4* d = (float4*)&Bt[0][stRow][stSeg * 8];
    d[0] = r0;
    d[1] = r1;
  }
  __syncthreads();

  for (int t = 0; t < TILES_PER_CHUNK; ++t) {
    const bool more = (t + 1 < TILES_PER_CHUNK);
    float4 r0, r1;
    if (more) {  // issue next tile's global loads early (overlap with WMMAs)
      const float4* g = (const float4*)(E +
          (size_t)(colChunk0 + (t + 1) * 16 + stRow) * DIM + stSeg * 8);
      r0 = g[0];
      r1 = g[1];
    }

    // Batch-read this tile's 32 B fragments from LDS (conflict-free b64).
    const float* bb = &Bt[t & 1][l16][koff];
    v2f bF[KSTEPS];
#pragma unroll
    for (int k = 0; k < KSTEPS; ++k) {
      bF[k].x = bb[4 * k];
      bF[k].y = bb[4 * k + 1];
    }

    const int col = colChunk0 + t * 16 + l16;
    const float sqB = sq[col];
    const int labB = labels[col];

    v8f acc0 = {};
    v8f acc1 = {};
#pragma unroll
    for (int k = 0; k < KSTEPS; k += 2) {
      acc0 = __builtin_amdgcn_wmma_f32_16x16x4_f32(false, aF[k], false, bF[k],
                                                   (short)0, acc0, false, false);
      acc1 = __builtin_amdgcn_wmma_f32_16x16x4_f32(false, aF[k + 1], false,
                                                   bF[k + 1], (short)0, acc1,
                                                   false, false);
    }

    const int colRel = col - rowBase - 8 * half;  // slot i is "self" iff == i
#pragma unroll
    for (int i = 0; i < 8; ++i) {
      const float dot = acc0[i] + acc1[i];
      const float d2 = (sqA[i] + sqB) - 2.0f * dot;
      const float c2 = fmaxf(d2, 0.0f);  // sqrt monotone: select on c2
      const bool same = (labA[i] == labB);
      const bool self = (colRel == i);
      if (same && !self && c2 > posv[i]) { posv[i] = c2; posi[i] = col; }
      if (!same && c2 < negv[i]) { negv[i] = c2; negi[i] = col; }
    }

    if (more) {  // store staged data into the other LDS buffer
      float4* d = (float4*)&Bt[(t + 1) & 1][stRow][stSeg * 8];
      d[0] = r0;
      d[1] = r1;
    }
    __syncthreads();
  }

  // Reduce across the 16 lanes of each half (xor masks 1..8 stay in-half).
#pragma unroll
  for (int m = 1; m < 16; m <<= 1) {
#pragma unroll
    for (int i = 0; i < 8; ++i) {
      float ov = __shfl_xor(posv[i], m, 32);
      int oi = __shfl_xor(posi[i], m, 32);
      if (ov > posv[i] || (ov == posv[i] && oi < posi[i])) { posv[i] = ov; posi[i] = oi; }
      ov = __shfl_xor(negv[i], m, 32);
      oi = __shfl_xor(negi[i], m, 32);
      if (ov < negv[i] || (ov == negv[i] && oi < negi[i])) { negv[i] = ov; negi[i] = oi; }
    }
  }

  // Lane 0 owns rows rowBase+0..7, lane 16 owns rows rowBase+8..15.
  if (l16 == 0) {
    const size_t base = (size_t)chunk * N_EMB + rowBase + 8 * half;
#pragma unroll
    for (int i = 0; i < 8; ++i) {
      cpv[base + i] = posv[i];
      cpi[base + i] = posi[i];
      cnv[base + i] = negv[i];
      cni[base + i] = negi[i];
    }
  }
}

// ---------------- Kernel 2b: combine column chunks per row ----------------
__global__ __launch_bounds__(256) void tl_combine(
    const float* __restrict__ cpv, const int* __restrict__ cpi,
    const float* __restrict__ cnv, const int* __restrict__ cni,
    int* __restrict__ posIdx, int* __restrict__ negIdx,
    float* __restrict__ validF) {
  const int r = blockIdx.x * 256 + threadIdx.x;
  float pv = cpv[r];
  int pi = cpi[r];
  float nv = cnv[r];
  int ni = cni[r];
#pragma unroll
  for (int c = 1; c < CHUNKS; ++c) {
    const size_t idx = (size_t)c * N_EMB + r;
    float ov = cpv[idx];
    int oi = cpi[idx];
    if (ov > pv || (ov == pv && oi < pi)) { pv = ov; pi = oi; }
    ov = cnv[idx];
    oi = cni[idx];
    if (ov < nv || (ov == nv && oi < ni)) { nv = ov; ni = oi; }
  }
  posIdx[r] = pi;
  negIdx[r] = ni;
  validF[r] = (pv > -INFINITY && nv < INFINITY) ? 1.0f : 0.0f;
}

// ---------------- Kernel 3: per-anchor hinge loss ----------------
__global__ __launch_bounds__(256) void tl_triplet(
    const float* __restrict__ E, const int* __restrict__ posIdx,
    const int* __restrict__ negIdx, const float* __restrict__ validF,
    float* __restrict__ perLoss) {
  const int wave = threadIdx.x >> 5;
  const int lane = threadIdx.x & 31;
  const int a = blockIdx.x * 8 + wave;
  const int p = posIdx[a];
  const int n = negIdx[a];
  const float4* A = (const float4*)(E + (size_t)a * DIM);
  const float4* P = (const float4*)(E + (size_t)p * DIM);
  const float4* G = (const float4*)(E + (size_t)n * DIM);
  const float4 av = A[lane], pv = P[lane], gv = G[lane];
  float dap = 0.0f, dan = 0.0f, d;
  d = av.x - pv.x + EPS_F; dap += d * d;
  d = av.y - pv.y + EPS_F; dap += d * d;
  d = av.z - pv.z + EPS_F; dap += d * d;
  d = av.w - pv.w + EPS_F; dap += d * d;
  d = av.x - gv.x + EPS_F; dan += d * d;
  d = av.y - gv.y + EPS_F; dan += d * d;
  d = av.z - gv.z + EPS_F; dan += d * d;
  d = av.w - gv.w + EPS_F; dan += d * d;
#pragma unroll
  for (int m = 1; m < 32; m <<= 1) {
    dap += __shfl_xor(dap, m, 32);
    dan += __shfl_xor(dan, m, 32);
  }
  if (lane == 0) {
    const float per = fmaxf(sqrtf(dap) - sqrtf(dan) + MARGIN_F, 0.0f);
    perLoss[a] = validF[a] * per;
  }
}

// ---------------- Kernel 4: deterministic final reduction ----------------
__global__ __launch_bounds__(256) void tl_finalize(
    const float* __restrict__ perLoss, const float* __restrict__ validF,
    float* __restrict__ out) {
  __shared__ float sT[256], sC[256];
  const int tid = threadIdx.x;
  float t = 0.0f, c = 0.0f;
  const int per = N_EMB / 256;  // fixed order per thread
  for (int j = 0; j < per; ++j) {
    const int i = tid * per + j;
    t += perLoss[i];
    c += validF[i];
  }
  sT[tid] = t;
  sC[tid] = c;
  __syncthreads();
  for (int s = 128; s > 0; s >>= 1) {
    if (tid < s) {
      sT[tid] += sT[tid + s];
      sC[tid] += sC[tid + s];
    }
    __syncthreads();
  }
  if (tid == 0) out[0] = (sC[0] > 0.0f) ? sT[0] / fmaxf(sC[0], 1.0f) : 0.0f;
}

extern "C" void kernel_launch(void* const* d_in, const int* in_sizes, int n_in,
                              void* d_out, int out_size, void* d_ws, size_t ws_size,
                              hipStream_t stream) {
  const float* E = (const float*)d_in[0];   // [N, D] fp32
  const int* labels = (const int*)d_in[1];  // [N] int
  float* out = (float*)d_out;               // scalar fp32

  // Workspace layout.
  float* sq = (float*)d_ws;                       // N
  float* cpv = sq + N_EMB;                        // CHUNKS*N
  int* cpi = (int*)(cpv + (size_t)CHUNKS * N_EMB);   // CHUNKS*N
  float* cnv = (float*)(cpi + (size_t)CHUNKS * N_EMB);  // CHUNKS*N
  int* cni = (int*)(cnv + (size_t)CHUNKS * N_EMB);   // CHUNKS*N
  int* posIdx = (int*)(cni + (size_t)CHUNKS * N_EMB);  // N
  int* negIdx = posIdx + N_EMB;                   // N
  float* perLoss = (float*)(negIdx + N_EMB);      // N
  float* validF = perLoss + N_EMB;                // N

  tl_rownorm<<<N_EMB / 8, 256, 0, stream>>>(E, sq);
  tl_gram_hard<<<ROWBLKS * CHUNKS, 256, 0, stream>>>(E, labels, sq, cpv, cpi,
                                                     cnv, cni);
  tl_combine<<<N_EMB / 256, 256, 0, stream>>>(cpv, cpi, cnv, cni, posIdx,
                                              negIdx, validF);
  tl_triplet<<<N_EMB / 8, 256, 0, stream>>>(E, posIdx, negIdx, validF, perLoss);
  tl_finalize<<<1, 256, 0, stream>>>(perLoss, validF, out);
}